// Regression_82403242541676
// MI455X (gfx1250) — compile-verified
//
#include <hip/hip_runtime.h>
#include <hip/hip_bf16.h>

// Problem dims (fixed by the reference)
#define BATCH 4
#define DISP  48
#define LH    128
#define LW    240
#define HH    512
#define HW    960

#if __has_builtin(__builtin_amdgcn_global_load_async_to_lds_b32)
#define HAVE_ASYNC_LDS 1
#else
#define HAVE_ASYNC_LDS 0
#endif

typedef __attribute__((address_space(1))) int* as1_i32p;
typedef __attribute__((address_space(3))) int* as3_i32p;

// ---------------------------------------------------------------------------
// Stage 1: per low-res pixel top-2 over disparity + 2-way softmax soft-argmax.
// Branch-free top-2 update -> v_cmp/v_cndmask chains, 48 loads pipelined.
// Grid is sized exactly (4*128*240 = 480*256), no tail guard needed.
// ---------------------------------------------------------------------------
__global__ __launch_bounds__(256) void topk_softargmax_kernel(
    const float* __restrict__ cost, float* __restrict__ disp4)
{
    const int idx = blockIdx.x * 256 + threadIdx.x;
    const int b   = idx / (LH * LW);
    const int rem = idx - b * (LH * LW);          // y*LW + x
    const float* p = cost + (size_t)b * DISP * LH * LW + rem;

    float v1 = p[0];                              // d = 0
    float v2 = -INFINITY;
    int   i1 = 0, i2 = 0;
    // Strict '>' matches jax.lax.top_k tie-breaking (earliest index wins).
#pragma unroll
    for (int d = 1; d < DISP; ++d) {
        const float v   = p[(size_t)d * (LH * LW)];
        const bool  gt1 = v > v1;
        const bool  gt2 = v > v2;
        const float nv2 = gt1 ? v1 : (gt2 ? v : v2);
        const int   ni2 = gt1 ? i1 : (gt2 ? d : i2);
        v1 = gt1 ? v : v1;
        i1 = gt1 ? d : i1;
        v2 = nv2;
        i2 = ni2;
    }
    // softmax([v1,v2]) dot [i1,i2], numerically stable since v2 <= v1.
    const float e2 = expf(v2 - v1);
    disp4[idx] = ((float)i1 + (float)i2 * e2) / (1.0f + e2);
}

// ---------------------------------------------------------------------------
// Stage 2: upfeat. Block = 64(X) x 4(Y) output pixels = 16 low-res cols x 1
// low-res row. Needed disp4 patch = 3 rows x 18 cols, staged in LDS via
// gfx1250 async global->LDS loads (ASYNCcnt). spg is streamed coalesced.
// ---------------------------------------------------------------------------
__global__ __launch_bounds__(256) void upfeat_kernel(
    const float* __restrict__ spg, const float* __restrict__ disp4,
    float* __restrict__ out)
{
    __shared__ float tile[3][20];                 // 18 used, padded

    const int b  = blockIdx.z;
    const int by = blockIdx.y;                    // low-res row index
    const int bx = blockIdx.x;                    // covers 16 low-res cols
    const int tx = threadIdx.x;                   // 0..63
    const int ty = threadIdx.y;                   // 0..3
    const int tid = ty * 64 + tx;

    // Stage the 3x18 disp4 patch (rows by-1..by+1, cols bx*16-1..bx*16+16).
    // Clamp OOB coords; validity is applied as a multiplier at consume time.
    if (tid < 54) {
        int row = tid / 18;
        int col = tid - row * 18;
        int gy = by + row - 1;       gy = gy < 0 ? 0 : (gy >= LH ? LH - 1 : gy);
        int gx = bx * 16 - 1 + col;  gx = gx < 0 ? 0 : (gx >= LW ? LW - 1 : gx);
        const float* src = disp4 + ((size_t)b * LH + gy) * LW + gx;
#if HAVE_ASYNC_LDS
        __builtin_amdgcn_global_load_async_to_lds_b32(
            (as1_i32p)(float*)src,
            (as3_i32p)&tile[row][col],
            0, 0);
#else
        tile[row][col] = *src;
#endif
    }
#if HAVE_ASYNC_LDS
#if __has_builtin(__builtin_amdgcn_s_wait_asynccnt)
    __builtin_amdgcn_s_wait_asynccnt(0);
#else
    asm volatile("s_wait_asynccnt 0" ::: "memory");
#endif
#endif
    __syncthreads();

    const int lx = tx >> 2;                       // low-res col within tile 0..15
    const bool rv0 = (by - 1) >= 0;               // dy==0 row valid
    const bool rv2 = (by + 1) < LH;               // dy==2 row valid
    const bool cv0 = (bx * 16 + lx - 1) >= 0;     // dx==0 col valid
    const bool cv2 = (bx * 16 + lx + 1) < LW;     // dx==2 col valid

    float pv[9];
#pragma unroll
    for (int dy = 0; dy < 3; ++dy) {
        const bool rv = (dy == 0) ? rv0 : ((dy == 2) ? rv2 : true);
#pragma unroll
        for (int dx = 0; dx < 3; ++dx) {
            const bool cv = (dx == 0) ? cv0 : ((dx == 2) ? cv2 : true);
            const float v = tile[dy][lx + dx];
            pv[dy * 3 + dx] = (rv && cv) ? v : 0.0f;
        }
    }

    const int Y = by * 4 + ty;
    const int X = bx * 64 + tx;
    const size_t spg_base = ((size_t)b * 9 * HH + Y) * (size_t)HW + X;

    float acc = 0.0f;
#pragma unroll
    for (int k = 0; k < 9; ++k) {
        acc = fmaf(spg[spg_base + (size_t)k * HH * HW], pv[k], acc);
    }
    out[((size_t)b * HH + Y) * (size_t)HW + X] = acc * 4.0f;
}

// ---------------------------------------------------------------------------
extern "C" void kernel_launch(void* const* d_in, const int* in_sizes, int n_in,
                              void* d_out, int out_size, void* d_ws, size_t ws_size,
                              hipStream_t stream) {
    const float* cost = (const float*)d_in[0];   // [4,1,48,128,240]
    const float* spg  = (const float*)d_in[1];   // [4,9,512,960]
    float* out   = (float*)d_out;                // [4,512,960]
    float* disp4 = (float*)d_ws;                 // [4,128,240] scratch

    const int n1 = BATCH * LH * LW;              // 122880 = 480 * 256 exactly
    topk_softargmax_kernel<<<n1 / 256, 256, 0, stream>>>(cost, disp4);

    dim3 grid(HW / 64, HH / 4, BATCH);
    dim3 block(64, 4, 1);
    upfeat_kernel<<<grid, block, 0, stream>>>(spg, disp4, out);
}